// LearnablePositionalOffset_22582938042481
// MI455X (gfx1250) — compile-verified
//
#include <hip/hip_runtime.h>
#include <math.h>

// LearnablePositionalOffset resample for MI455X (gfx1250).
//
// Roofline: 512 MB stores + 64 MB reads @ 23.3 TB/s  => ~25 us floor.
// Pure memory-bound gather; no matmul structure => WMMA inapplicable
// (dense-matmul formulation would need ~2.7 PF/s of f32 WMMA to match).
// CDNA5 path: TENSOR_LOAD_TO_LDS (TDM) stages each 256 KB (b,c) plane
// into LDS once (HBM reads at the 64 MB floor), gathers are conflict-free
// ds_load_b32, output is coalesced non-temporal stores.
//
// Issue-rate budget per pixel (steady state): ~2.25 ds + ~7 VALU + 1 vmem,
// via row-value reuse (y0(h+1)==y1(h) common case) and per-thread hoisted
// x-side LUT/addressing.

#define H_ 256
#define W_ 256
#define B_ 8
#define C_ 32
#define K_ 8

typedef unsigned int v4u __attribute__((ext_vector_type(4)));
typedef int          v8i __attribute__((ext_vector_type(8)));
typedef int          v4i __attribute__((ext_vector_type(4)));

#ifdef __has_builtin
#if __has_builtin(__builtin_amdgcn_tensor_load_to_lds)
#define USE_TDM 1
#endif
#endif
#ifndef USE_TDM
#define USE_TDM 0
#endif

// Replicates the reference coordinate transform + reflect in fp32.
// Returns {bitcast(i0), frac}.
__device__ __forceinline__ float2 coord_lut(int p, float d, float size) {
    float nx = 2.0f * ((float)p + d) / (size - 1.0f) - 1.0f;
    float ix = ((nx + 1.0f) * size - 1.0f) * 0.5f;
    // _reflect
    const float m = -0.5f;
    float cc    = fabsf(ix - m);
    float extra = fmodf(cc, size);
    float flips = floorf(cc / size);
    float r = (fmodf(flips, 2.0f) == 0.0f) ? (extra + m) : (size - extra + m);
    r = fminf(fmaxf(r, 0.0f), size - 1.0f);
    float f0 = floorf(r);
    float fr = r - f0;
    int   i0 = min(max((int)f0, 0), (int)size - 1);
    float2 res;
    res.x = __int_as_float(i0);
    res.y = fr;
    return res;
}

__global__ __launch_bounds__(1024, 1)
void lpo_tdm_kernel(const float* __restrict__ x,
                    const float* __restrict__ offx,
                    const float* __restrict__ offy,
                    const float* __restrict__ scale,
                    float* __restrict__ out) {
    extern __shared__ unsigned char smem_raw[];
    float*  plane = (float*)smem_raw;                               // 256 KB: one (b,c) plane
    float2* lutX  = (float2*)(smem_raw + (size_t)H_ * W_ * 4);      // 16 KB: {x0i, wx}[k][w]
    float2* lutY  = lutX + K_ * W_;                                 // 16 KB: {y0i, wy}[k][h]

    const int tid = threadIdx.x;
    const int bc  = blockIdx.x;
    const int b   = bc >> 5;
    const int c   = bc & 31;
    const float* src = x + (size_t)(b * C_ + c) * (H_ * W_);

    // ---- Stage the full plane into LDS via the Tensor Data Mover ----
#if USE_TDM
    if (tid < 32) {  // wave 0 only (TDM ignores EXEC; one issue per block)
        unsigned lds_off = (unsigned)(unsigned long long)(void*)plane; // low 32b = LDS byte offset
        unsigned long long ga = (unsigned long long)(const void*)src;
        v4u g0;
        g0[0] = 1u;                                   // count=1 valid descriptor, not restore
        g0[1] = lds_off;                              // lds_addr [63:32]
        g0[2] = (unsigned)(ga & 0xffffffffull);       // global_addr [95:64]
        g0[3] = (unsigned)((ga >> 32) & 0x01ffffffull) | (2u << 30); // addr[56:32] | type=2
        v8i g1;
        g1[0] = 0x00020000;                           // data_size=2 (4 bytes)
        g1[1] = (int)(256u << 16);                    // tensor_dim0 = 256   (bits 63:48)
        g1[2] = (int)(256u << 16);                    // tensor_dim1 = 256   (bits 95:80)
        g1[3] = (int)(256u << 16);                    // tile_dim0   = 256   (bits 127:112)
        g1[4] = 256;                                  // tile_dim1   = 256   (bits 143:128)
        g1[5] = 256;                                  // tensor_dim0_stride  (bits 191:160)
        g1[6] = 0;                                    // stride0 hi / stride1 lo16
        g1[7] = 1;                                    // tensor_dim1_stride = 65536 (bits 255:224)
        v4i gz4 = {0, 0, 0, 0};                       // groups 2/3 unused (2D tensor)
        v8i gz8 = {0, 0, 0, 0, 0, 0, 0, 0};          // trailing group (clang-23 6-arg form)
        __builtin_amdgcn_tensor_load_to_lds(g0, g1, gz4, gz4, gz8, 0);
    }
#else
    for (int i = tid; i < H_ * W_ / 4; i += 1024) {
        ((float4*)plane)[i] = ((const float4*)src)[i];
    }
#endif

    // ---- Build tiny per-(k,coord) LUTs while the DMA is in flight ----
    const float s = scale[0] * 16.0f;  // scale * MAX_OFFSET
    for (int e = tid; e < K_ * 256; e += 1024) {
        int k = e >> 8;
        int p = e & 255;
        lutX[e] = coord_lut(p, offx[k] * s, 256.0f);
        lutY[e] = coord_lut(p, offy[k] * s, 256.0f);
    }

#if USE_TDM
    if (tid < 32) __builtin_amdgcn_s_wait_tensorcnt(0);
#endif
    __syncthreads();

    // ---- Gather phase -------------------------------------------------
    // Thread owns: one offset k, 4 columns w = q + 64j (lanes -> consecutive
    // w: conflict-free LDS banks, coalesced stores), and a sequential half
    // of the h range so x-lerped row values can be reused across rows.
    const int k    = tid >> 7;        // 0..7
    const int rest = tid & 127;
    const int q    = rest & 63;       // column group
    const int hh   = rest >> 6;       // 0..1 : which half of h

    int   xo0[4], xo1[4];
    float wxv[4];
#pragma unroll
    for (int j = 0; j < 4; ++j) {
        float2 lx = lutX[(k << 8) + q + 64 * j];
        int x0 = __float_as_int(lx.x);
        xo0[j] = x0;
        xo1[j] = min(x0 + 1, W_ - 1);
        wxv[j] = lx.y;
    }
    float* obase = out + (size_t)((b * K_ + k) * C_ + c) * (H_ * W_);

    float rv_prev[4];
    int   y_prev = -1;                // no cached row yet
    const int h0 = hh << 7;
    for (int h = h0; h < h0 + 128; ++h) {
        float2 ly = lutY[(k << 8) + h];
        int   y0 = __float_as_int(ly.x);
        float wy = ly.y;
        int   y1 = min(y0 + 1, H_ - 1);

        float rv0[4];
        if (y0 == y_prev) {           // wave-uniform: cheap execz skip
#pragma unroll
            for (int j = 0; j < 4; ++j) rv0[j] = rv_prev[j];
        } else {
            const float* r0 = plane + (y0 << 8);
#pragma unroll
            for (int j = 0; j < 4; ++j) {
                float a = r0[xo0[j]];
                float e = r0[xo1[j]];
                rv0[j] = a + wxv[j] * (e - a);
            }
        }

        const float* r1 = plane + (y1 << 8);
        float rv1[4];
#pragma unroll
        for (int j = 0; j < 4; ++j) {
            float a = r1[xo0[j]];
            float e = r1[xo1[j]];
            rv1[j] = a + wxv[j] * (e - a);
        }

        float* orow = obase + (h << 8) + q;
#pragma unroll
        for (int j = 0; j < 4; ++j) {
            float o = rv0[j] + wy * (rv1[j] - rv0[j]);
            __builtin_nontemporal_store(o, orow + 64 * j);
            rv_prev[j] = rv1[j];
        }
        y_prev = y1;
    }
}

extern "C" void kernel_launch(void* const* d_in, const int* in_sizes, int n_in,
                              void* d_out, int out_size, void* d_ws, size_t ws_size,
                              hipStream_t stream) {
    (void)in_sizes; (void)n_in; (void)d_ws; (void)ws_size; (void)out_size;
    const float* x     = (const float*)d_in[0];
    const float* offx  = (const float*)d_in[1];
    const float* offy  = (const float*)d_in[2];
    const float* scale = (const float*)d_in[3];
    float* out = (float*)d_out;

    // 256 KB plane + 2 * 16 KB LUTs = 288 KB dynamic LDS (<= 320 KB/WGP)
    size_t shmem = (size_t)H_ * W_ * sizeof(float) + 2u * K_ * 256 * sizeof(float2);
    lpo_tdm_kernel<<<dim3(B_ * C_), dim3(1024), shmem, stream>>>(x, offx, offy, scale, out);
}